// Cell_67894843015282
// MI455X (gfx1250) — compile-verified
//
#include <hip/hip_runtime.h>
#include <cmath>

typedef __attribute__((ext_vector_type(2))) float v2f;
typedef __attribute__((ext_vector_type(8))) float v8f;

#define HWSZ 1024   // 32*32
#define NCH  128

// ------------------------------------------------------------------
// 1-thread kernel: sequential edge/op mask recurrence -> coef[e,k]=w*m
// ------------------------------------------------------------------
__global__ void k_coef(const float* __restrict__ w2,
                       const float* __restrict__ thre,
                       const int*   __restrict__ mdef,
                       float* __restrict__ coef)
{
  if (threadIdx.x || blockIdx.x) return;
  int offset = 0;
  for (int i = 0; i < 4; ++i) {
    int n = 2 + i;
    float wsum = 0.f;
    for (int j = 0; j < n; ++j)
      for (int k = 0; k < 8; ++k)
        wsum += w2[(offset + j) * 8 + k] * (float)mdef[(offset + j) * 8 + k];
    for (int j = 0; j < n; ++j) {
      int e = offset + j;
      float ns = wsum, msum = 0.f, dsum = 0.f;
      for (int k = 0; k < 8; ++k) {
        float w  = w2[e * 8 + k];
        float md = (float)mdef[e * 8 + k];
        float t0 = thre[(e * 8 + k) * 3 + 0];
        float m;
        if (md == 0.f)    m = 0.f;
        else if (w != ns) m = (w - t0) > 0.f ? 1.f : 0.f;
        else              m = md;
        bool cond = (md != 0.f) && (w != ns) && (m == 0.f);
        coef[e * 8 + k] = w * m;
        msum += w * m;
        dsum += w * md;
        if (cond) ns -= w;
      }
      wsum = wsum - dsum + msum;
    }
    offset += n;
  }
}

// ------------------------------------------------------------------
// preprocess channel affine: A = g * gate*mk ; B = b * gate*mk
// layout pre[2][2][128] : [i][0]=A, [i][1]=B
// ------------------------------------------------------------------
__global__ void k_chan_pre(const float* __restrict__ kp, const float* __restrict__ tp,
                           const float* __restrict__ g0, const float* __restrict__ b0,
                           const float* __restrict__ g1, const float* __restrict__ b1,
                           float* __restrict__ pre)
{
  int i = blockIdx.x, c = threadIdx.x;
  float gate = 1.f / (1.f + expf(-kp[i * NCH + c]));
  float mk = (gate - tp[i]) > 0.f ? 1.f : 0.f;
  float s = gate * mk;
  const float* g = i ? g1 : g0;
  const float* b = i ? b1 : b0;
  pre[i * 256 + c]       = g[c] * s;
  pre[i * 256 + 128 + c] = b[c] * s;
}

// ------------------------------------------------------------------
// per-edge channel affines for sep3/sep5/dil3/dil5.
// chp layout: [14][12][128] slots:
//  0:s3A1 1:s3B1 2:s3A2 3:s3B2 4:s5A1 5:s5B1 6:s5A2 7:s5B2 8:d3A 9:d3B 10:d5A 11:d5B
// ------------------------------------------------------------------
__global__ void k_chan_edges(const float* __restrict__ kparam, const float* __restrict__ thre,
                             const int* __restrict__ mkd, const int* __restrict__ mwd,
                             const float* __restrict__ s3g1, const float* __restrict__ s3b1,
                             const float* __restrict__ s3g2, const float* __restrict__ s3b2,
                             const float* __restrict__ s5g1, const float* __restrict__ s5b1,
                             const float* __restrict__ s5g2, const float* __restrict__ s5b2,
                             const float* __restrict__ d3g,  const float* __restrict__ d3b,
                             const float* __restrict__ d5g,  const float* __restrict__ d5b,
                             float* __restrict__ chp)
{
  int e = blockIdx.x >> 2, q = blockIdx.x & 3, k = 4 + q, c = threadIdx.x;
  int ek = e * 8 + k;
  float gate = 1.f / (1.f + expf(-kparam[ek * NCH + c]));
  float t1 = thre[ek * 3 + 1], t2 = thre[ek * 3 + 2];
  float mk = ((gate - t1) > 0.f && mkd[ek * NCH + c] != 0) ? 1.f : 0.f;
  float* base = chp + e * 12 * NCH;
  if (q < 2) {
    float mw = ((gate - t2) > 0.f && mwd[ek * NCH + c] != 0) ? 1.f : 0.f;
    float s1 = gate * mk, s2 = gate * mw;
    const float* g1 = q ? s5g1 : s3g1; const float* bb1 = q ? s5b1 : s3b1;
    const float* g2 = q ? s5g2 : s3g2; const float* bb2 = q ? s5b2 : s3b2;
    int off = q * 4 * NCH;
    base[off + 0 * NCH + c] = g1[e * NCH + c] * s1;
    base[off + 1 * NCH + c] = bb1[e * NCH + c] * s1;
    base[off + 2 * NCH + c] = g2[e * NCH + c] * s2;
    base[off + 3 * NCH + c] = bb2[e * NCH + c] * s2;
  } else {
    float s = gate * mk;
    const float* g = (q == 2) ? d3g : d5g;
    const float* b = (q == 2) ? d3b : d5b;
    int off = 8 * NCH + (q - 2) * 2 * NCH;
    base[off + c]       = g[e * NCH + c] * s;
    base[off + NCH + c] = b[e * NCH + c] * s;
  }
}

// ------------------------------------------------------------------
// WMMA f32 GEMM: per batch  Out[128,1024] = W[128,K] x X[K,1024]
// then per-channel affine y = A[m]*dot + B[m];
// ACC==0 : Out = y ; ACC==1 : Out += (*coefp) * y
// RELU   : apply relu to X elements on load (preprocess only)
// block = 128 threads (4 waves); wave -> 16(M) x 64(N) tile via
// v_wmma_f32_16x16x4_f32, K-loop step 4.  grid = (128/64, 1024/64, 8)
// ------------------------------------------------------------------
template <int RELU, int ACC>
__global__ __launch_bounds__(128)
void k_gemm(const float* __restrict__ X, long xb,
            const float* __restrict__ W, int K,
            const float* __restrict__ Asc, const float* __restrict__ Bbs,
            float* __restrict__ O, long ob,
            const float* __restrict__ coefp)
{
  const int lane = threadIdx.x & 31;
  const int wave = threadIdx.x >> 5;
  const int half = lane >> 4;
  const int lm   = lane & 15;
  const int mb = blockIdx.x * 64 + wave * 16;
  const int nb = blockIdx.y * 64;
  const float* Xb = X + (long)blockIdx.z * xb;

  v8f acc0 = {}, acc1 = {}, acc2 = {}, acc3 = {};

  const float* wrow = W + (mb + lm) * K + 2 * half;           // A: row m, K = 2*half + {0,1}
  const float* xcol = Xb + (long)(2 * half) * HWSZ + nb + lm; // B: K = 2*half + {0,1}, col n

  for (int k0 = 0; k0 < K; k0 += 4) {
    v2f a;
    a[0] = wrow[0];
    a[1] = wrow[1];
    float b00 = xcol[0],  b01 = xcol[HWSZ];
    float b10 = xcol[16], b11 = xcol[16 + HWSZ];
    float b20 = xcol[32], b21 = xcol[32 + HWSZ];
    float b30 = xcol[48], b31 = xcol[48 + HWSZ];
    if (RELU) {
      b00 = fmaxf(b00, 0.f); b01 = fmaxf(b01, 0.f);
      b10 = fmaxf(b10, 0.f); b11 = fmaxf(b11, 0.f);
      b20 = fmaxf(b20, 0.f); b21 = fmaxf(b21, 0.f);
      b30 = fmaxf(b30, 0.f); b31 = fmaxf(b31, 0.f);
    }
    v2f bb0 = {b00, b01}, bb1 = {b10, b11}, bb2 = {b20, b21}, bb3 = {b30, b31};
    acc0 = __builtin_amdgcn_wmma_f32_16x16x4_f32(false, a, false, bb0, (short)0, acc0, false, false);
    acc1 = __builtin_amdgcn_wmma_f32_16x16x4_f32(false, a, false, bb1, (short)0, acc1, false, false);
    acc2 = __builtin_amdgcn_wmma_f32_16x16x4_f32(false, a, false, bb2, (short)0, acc2, false, false);
    acc3 = __builtin_amdgcn_wmma_f32_16x16x4_f32(false, a, false, bb3, (short)0, acc3, false, false);
    wrow += 4;
    xcol += 4 * HWSZ;
  }

  const float cf = ACC ? coefp[0] : 0.f;
  #pragma unroll
  for (int v = 0; v < 8; ++v) {
    int m = mb + v + 8 * half;                 // D: VGPR v -> row m = v + 8*half
    float A = Asc[m], Bb = Bbs[m];
    float* orow = O + (long)blockIdx.z * ob + (long)m * HWSZ + nb + lm;
    float y0 = A * acc0[v] + Bb;
    float y1 = A * acc1[v] + Bb;
    float y2 = A * acc2[v] + Bb;
    float y3 = A * acc3[v] + Bb;
    if (ACC) {
      orow[0]  += cf * y0; orow[16] += cf * y1;
      orow[32] += cf * y2; orow[48] += cf * y3;
    } else {
      orow[0]  = y0; orow[16] = y1; orow[32] = y2; orow[48] = y3;
    }
  }
}

// ------------------------------------------------------------------
// depthwise conv (groups=C), SAME padding, ReLU applied to input
// ------------------------------------------------------------------
template <int KS, int DIL>
__global__ __launch_bounds__(256)
void k_dw(const float* __restrict__ X, long xb,
          const float* __restrict__ Wd, float* __restrict__ O)
{
  int idx = blockIdx.x * 256 + threadIdx.x;       // 8*128*1024 threads
  int hw = idx & (HWSZ - 1);
  int c  = (idx >> 10) & (NCH - 1);
  int b  = idx >> 17;
  int h = hw >> 5, w = hw & 31;
  const float* xp = X + (long)b * xb + (long)c * HWSZ;
  const float* wp = Wd + c * KS * KS;
  const int pad = DIL * (KS - 1) / 2;
  float s = 0.f;
  #pragma unroll
  for (int ky = 0; ky < KS; ++ky) {
    int yy = h + DIL * ky - pad;
    if ((unsigned)yy < 32u) {
      #pragma unroll
      for (int kx = 0; kx < KS; ++kx) {
        int xx = w + DIL * kx - pad;
        if ((unsigned)xx < 32u)
          s += wp[ky * KS + kx] * fmaxf(xp[yy * 32 + xx], 0.f);
      }
    }
  }
  O[(long)b * (NCH * HWSZ) + (long)c * HWSZ + hw] = s;
}

// ------------------------------------------------------------------
// fused maxpool3 / avgpool3(valid-count) / skip:
//   acc (=|+=) c1*maxp + c2*avgp + c3*x
// ------------------------------------------------------------------
template <int INIT>
__global__ __launch_bounds__(256)
void k_pool_skip(const float* __restrict__ X, long xb,
                 float* __restrict__ Acc, long ab,
                 const float* __restrict__ coef, int e)
{
  float c1 = coef[e * 8 + 1], c2 = coef[e * 8 + 2], c3 = coef[e * 8 + 3];
  int idx = blockIdx.x * 256 + threadIdx.x;
  int hw = idx & (HWSZ - 1);
  int c  = (idx >> 10) & (NCH - 1);
  int b  = idx >> 17;
  int h = hw >> 5, w = hw & 31;
  const float* xp = X + (long)b * xb + (long)c * HWSZ;
  float ctr = xp[h * 32 + w];
  float mx = ctr, sm = 0.f;
  int cnt = 0;
  for (int dy = -1; dy <= 1; ++dy) {
    int yy = h + dy;
    if ((unsigned)yy >= 32u) continue;
    for (int dx = -1; dx <= 1; ++dx) {
      int xx = w + dx;
      if ((unsigned)xx >= 32u) continue;
      float v = xp[yy * 32 + xx];
      mx = fmaxf(mx, v);
      sm += v;
      ++cnt;
    }
  }
  float val = c1 * mx + c2 * (sm / (float)cnt) + c3 * ctr;
  float* o = Acc + (long)b * ab + (long)c * HWSZ + hw;
  *o = INIT ? val : (*o + val);
}

// ------------------------------------------------------------------
extern "C" void kernel_launch(void* const* d_in, const int* in_sizes, int n_in,
                              void* d_out, int out_size, void* d_ws, size_t ws_size,
                              hipStream_t stream)
{
  (void)in_sizes; (void)n_in; (void)out_size; (void)ws_size;
  const float* s0        = (const float*)d_in[0];
  const float* s1        = (const float*)d_in[1];
  const float* pre0_w    = (const float*)d_in[2];
  const float* pre0_g    = (const float*)d_in[3];
  const float* pre0_b    = (const float*)d_in[4];
  const float* pre1_w    = (const float*)d_in[5];
  const float* pre1_g    = (const float*)d_in[6];
  const float* pre1_b    = (const float*)d_in[7];
  const float* kernel_pre= (const float*)d_in[8];
  const float* thre_pre  = (const float*)d_in[9];
  const float* weights2  = (const float*)d_in[10];
  const float* kernel_pa = (const float*)d_in[11];
  const float* thre      = (const float*)d_in[12];
  const float* s3dw1     = (const float*)d_in[13];
  const float* s3pw1     = (const float*)d_in[14];
  const float* s3g1      = (const float*)d_in[15];
  const float* s3b1      = (const float*)d_in[16];
  const float* s3dw2     = (const float*)d_in[17];
  const float* s3pw2     = (const float*)d_in[18];
  const float* s3g2      = (const float*)d_in[19];
  const float* s3b2      = (const float*)d_in[20];
  const float* s5dw1     = (const float*)d_in[21];
  const float* s5pw1     = (const float*)d_in[22];
  const float* s5g1      = (const float*)d_in[23];
  const float* s5b1      = (const float*)d_in[24];
  const float* s5dw2     = (const float*)d_in[25];
  const float* s5pw2     = (const float*)d_in[26];
  const float* s5g2      = (const float*)d_in[27];
  const float* s5b2      = (const float*)d_in[28];
  const float* d3dw      = (const float*)d_in[29];
  const float* d3pw      = (const float*)d_in[30];
  const float* d3g       = (const float*)d_in[31];
  const float* d3b       = (const float*)d_in[32];
  const float* d5dw      = (const float*)d_in[33];
  const float* d5pw      = (const float*)d_in[34];
  const float* d5g       = (const float*)d_in[35];
  const float* d5b       = (const float*)d_in[36];
  const int*   mask_def  = (const int*)d_in[37];
  const int*   mask_kdef = (const int*)d_in[38];
  const int*   mask_wdef = (const int*)d_in[39];

  float* ws   = (float*)d_ws;
  float* coef = ws;            // 112 floats
  float* pre  = ws + 128;      // 512 floats
  float* chp  = ws + 640;      // 14*12*128 = 21504
  float* h0   = ws + 22528;    // 8*128*1024 = 1048576
  float* h1   = h0 + 1048576;
  float* T1   = h1 + 1048576;
  float* T2   = T1 + 1048576;  // total ~16.9 MB
  float* out  = (float*)d_out;

  // scalar + channel parameter setup
  k_coef<<<1, 1, 0, stream>>>(weights2, thre, mask_def, coef);
  k_chan_pre<<<2, 128, 0, stream>>>(kernel_pre, thre_pre, pre0_g, pre0_b, pre1_g, pre1_b, pre);
  k_chan_edges<<<56, 128, 0, stream>>>(kernel_pa, thre, mask_kdef, mask_wdef,
                                       s3g1, s3b1, s3g2, s3b2, s5g1, s5b1, s5g2, s5b2,
                                       d3g, d3b, d5g, d5b, chp);

  dim3 gg(2, 16, 8);  // (M/64, 1024/64, B)
  // preprocess: h = (g*gate*mk)*conv1x1(relu(s)) + (b*gate*mk)
  k_gemm<1, 0><<<gg, 128, 0, stream>>>(s0, 512L * 1024, pre0_w, 512, pre,       pre + 128, h0, 131072L, nullptr);
  k_gemm<1, 0><<<gg, 128, 0, stream>>>(s1, 512L * 1024, pre1_w, 512, pre + 256, pre + 384, h1, 131072L, nullptr);

  struct St { const float* p; long bs; } st[6];
  st[0] = { h0, 131072L };
  st[1] = { h1, 131072L };
  for (int i = 0; i < 4; ++i)
    st[2 + i] = { out + (long)i * 131072, 524288L };   // concat layout in d_out

  int e = 0;
  for (int i = 0; i < 4; ++i) {
    int n = 2 + i;
    float* acc = (float*)st[2 + i].p;
    long   ab  = st[2 + i].bs;
    for (int j = 0; j < n; ++j, ++e) {
      const float* x = st[j].p;
      long xb = st[j].bs;
      float* ech = chp + e * 12 * 128;

      // k=1,2,3: pools + skip (first edge initializes the node accumulator)
      if (j == 0)
        k_pool_skip<1><<<4096, 256, 0, stream>>>(x, xb, acc, ab, coef, e);
      else
        k_pool_skip<0><<<4096, 256, 0, stream>>>(x, xb, acc, ab, coef, e);

      // k=4: sep3 = dw3 -> pw1(aff1) -> dw3(relu) -> pw2(aff2) accum
      k_dw<3, 1><<<4096, 256, 0, stream>>>(x, xb, s3dw1 + e * 1152, T1);
      k_gemm<0, 0><<<gg, 128, 0, stream>>>(T1, 131072L, s3pw1 + e * 16384, 128, ech + 0,   ech + 128, T2, 131072L, nullptr);
      k_dw<3, 1><<<4096, 256, 0, stream>>>(T2, 131072L, s3dw2 + e * 1152, T1);
      k_gemm<0, 1><<<gg, 128, 0, stream>>>(T1, 131072L, s3pw2 + e * 16384, 128, ech + 256, ech + 384, acc, ab, coef + e * 8 + 4);

      // k=5: sep5
      k_dw<5, 1><<<4096, 256, 0, stream>>>(x, xb, s5dw1 + e * 3200, T1);
      k_gemm<0, 0><<<gg, 128, 0, stream>>>(T1, 131072L, s5pw1 + e * 16384, 128, ech + 512, ech + 640, T2, 131072L, nullptr);
      k_dw<5, 1><<<4096, 256, 0, stream>>>(T2, 131072L, s5dw2 + e * 3200, T1);
      k_gemm<0, 1><<<gg, 128, 0, stream>>>(T1, 131072L, s5pw2 + e * 16384, 128, ech + 768, ech + 896, acc, ab, coef + e * 8 + 5);

      // k=6: dil3 (dilation=2)
      k_dw<3, 2><<<4096, 256, 0, stream>>>(x, xb, d3dw + e * 1152, T1);
      k_gemm<0, 1><<<gg, 128, 0, stream>>>(T1, 131072L, d3pw + e * 16384, 128, ech + 1024, ech + 1152, acc, ab, coef + e * 8 + 6);

      // k=7: dil5 (dilation=2)
      k_dw<5, 2><<<4096, 256, 0, stream>>>(x, xb, d5dw + e * 3200, T1);
      k_gemm<0, 1><<<gg, 128, 0, stream>>>(T1, 131072L, d5pw + e * 16384, 128, ech + 1280, ech + 1408, acc, ab, coef + e * 8 + 7);
    }
  }
}